// SchemaHeteroGAT_7722351198342
// MI455X (gfx1250) — compile-verified
//
#include <hip/hip_runtime.h>
#include <math.h>

// ---------------------------------------------------------------------------
// Types for WMMA fragments (wave32, gfx1250)
// ---------------------------------------------------------------------------
typedef float v2f __attribute__((ext_vector_type(2)));
typedef float v8f __attribute__((ext_vector_type(8)));

// ---------------------------------------------------------------------------
// Generic fp32 WMMA GEMM:  C[M,N] = act(A[M,K] @ W[K,N] + bias)   (or C += ...)
// act: 0 = none, 1 = leaky_relu(0.01)
// Requirements: K % 4 == 0, N % 64 == 0.  M arbitrary (row-clamped loads,
// row-guarded stores).
// Block = 256 threads = 8 waves; each wave computes one 32(M) x 64(N) tile
// (2 A fragments x 4 B fragments = 8 x V_WMMA_F32_16X16X4_F32 per k-step,
// exact fp32 matrix math). Inner loop is pure pointer-increment addressing,
// no divergence.
// ---------------------------------------------------------------------------
__global__ __launch_bounds__(256)
void gemm_wmma_f32(const float* __restrict__ A, const float* __restrict__ W,
                   const float* __restrict__ bias, float* __restrict__ C,
                   int M, int N, int K, int act, int accumulate)
{
    const int lane = threadIdx.x & 31;
    const int wave = threadIdx.x >> 5;
    const int g    = lane >> 4;       // half-wave group (0 or 1)
    const int ln   = lane & 15;

    const int numMt = (M + 31) >> 5;             // 32-row tiles
    const int mt    = blockIdx.x * 8 + wave;
    if (mt >= numMt) return;                     // wave-uniform exit
    const int n0    = blockIdx.y * 64;           // 64-wide N tile

    // A rows this lane feeds (clamped: duplicates compute garbage never stored)
    const int row0 = mt * 32 + ln;
    const int row1 = row0 + 16;
    const int r0c  = (row0 < M) ? row0 : (M - 1);
    const int r1c  = (row1 < M) ? row1 : (M - 1);

    const float* Ap0 = A + (size_t)r0c * K + 2 * g;   // 8B-aligned (K%4==0)
    const float* Ap1 = A + (size_t)r1c * K + 2 * g;
    const float* Wp  = W + (size_t)(2 * g) * N + n0 + ln;  // row kk=2g of B tile

    v8f acc[2][4] = {};                          // [m-subtile][n-subtile]

    for (int k = 0; k < K; k += 4) {
        v2f a0 = *(const v2f*)Ap0;               // A[r0][k+2g], A[r0][k+2g+1]
        v2f a1 = *(const v2f*)Ap1;
#pragma unroll
        for (int t = 0; t < 4; ++t) {
            v2f b;
            b.x = Wp[16 * t];                    // B[k+2g  ][n0+16t+ln]
            b.y = Wp[(size_t)N + 16 * t];        // B[k+2g+1][n0+16t+ln]
            acc[0][t] = __builtin_amdgcn_wmma_f32_16x16x4_f32(
                false, a0, false, b, (short)0, acc[0][t], false, false);
            acc[1][t] = __builtin_amdgcn_wmma_f32_16x16x4_f32(
                false, a1, false, b, (short)0, acc[1][t], false, false);
        }
        Ap0 += 4; Ap1 += 4; Wp += (size_t)4 * N;
    }

#pragma unroll
    for (int t = 0; t < 4; ++t) {
        const int col = n0 + 16 * t + ln;
        const float bv = bias ? bias[col] : 0.0f;
#pragma unroll
        for (int u = 0; u < 2; ++u) {
#pragma unroll
            for (int i = 0; i < 8; ++i) {
                const int m = mt * 32 + u * 16 + 8 * g + i;  // D VGPR i -> row i+8g
                if (m < M) {
                    float v = acc[u][t][i] + bv;
                    if (act == 1) v = (v > 0.0f) ? v : 0.01f * v;
                    float* cp = C + (size_t)m * N + col;
                    if (accumulate) *cp += v; else *cp = v;
                }
            }
        }
    }
}

// ---------------------------------------------------------------------------
// Utility fill
// ---------------------------------------------------------------------------
__global__ void fill_f32(float* __restrict__ p, float v, size_t n)
{
    size_t i = (size_t)blockIdx.x * blockDim.x + threadIdx.x;
    if (i < n) p[i] = v;
}

// ---------------------------------------------------------------------------
// Mixed-sign float atomic max (word interpreted as int for >=0, uint for <0).
// Buffer must be initialized to -inf (0xFF800000).
// ---------------------------------------------------------------------------
__device__ inline void atomicMaxF(float* addr, float v)
{
    if (v >= 0.0f)
        atomicMax((int*)addr, __float_as_int(v));
    else
        atomicMin((unsigned int*)addr, __float_as_uint(v));
}

// ---------------------------------------------------------------------------
// Per-edge attention logits + segment max.
// logit[e][h] = sum_c att[h*64+c] * leaky02(xl[src][h*64+c] + xr[dst][h*64+c])
// ---------------------------------------------------------------------------
__global__ void edge_logit_kernel(const int* __restrict__ src, const int* __restrict__ dst,
                                  const float* __restrict__ xl, const float* __restrict__ xr,
                                  const float* __restrict__ att,
                                  float* __restrict__ logit, float* __restrict__ mbuf, int E)
{
    int e = blockIdx.x * blockDim.x + threadIdx.x;
    if (e >= E) return;
    const int s = src[e], d = dst[e];
    const float* xls = xl + (size_t)s * 256;
    const float* xrd = xr + (size_t)d * 256;
    float a0 = 0.f, a1 = 0.f, a2 = 0.f, a3 = 0.f;
    for (int c = 0; c < 64; ++c) {
        float v;
        v = xls[c]       + xrd[c];       v = (v > 0.f) ? v : 0.2f * v; a0 += att[c]       * v;
        v = xls[64 + c]  + xrd[64 + c];  v = (v > 0.f) ? v : 0.2f * v; a1 += att[64 + c]  * v;
        v = xls[128 + c] + xrd[128 + c]; v = (v > 0.f) ? v : 0.2f * v; a2 += att[128 + c] * v;
        v = xls[192 + c] + xrd[192 + c]; v = (v > 0.f) ? v : 0.2f * v; a3 += att[192 + c] * v;
    }
    logit[e * 4 + 0] = a0;  atomicMaxF(mbuf + (size_t)d * 4 + 0, a0);
    logit[e * 4 + 1] = a1;  atomicMaxF(mbuf + (size_t)d * 4 + 1, a1);
    logit[e * 4 + 2] = a2;  atomicMaxF(mbuf + (size_t)d * 4 + 2, a2);
    logit[e * 4 + 3] = a3;  atomicMaxF(mbuf + (size_t)d * 4 + 3, a3);
}

// ---------------------------------------------------------------------------
// ex = exp(logit - m[dst]); den[dst] += ex  (logit overwritten with ex)
// ---------------------------------------------------------------------------
__global__ void edge_expsum_kernel(const int* __restrict__ dst,
                                   float* __restrict__ logit,
                                   const float* __restrict__ mbuf,
                                   float* __restrict__ den, int E)
{
    int i = blockIdx.x * blockDim.x + threadIdx.x;
    if (i >= E * 4) return;
    const int e = i >> 2, h = i & 3;
    const int d = dst[e];
    float ex = expf(logit[i] - mbuf[(size_t)d * 4 + h]);
    logit[i] = ex;
    atomicAdd(den + (size_t)d * 4 + h, ex);
}

// ---------------------------------------------------------------------------
// Weighted message scatter: acc[dst][c] += xl[src][c] * ex[e][h]/den[dst][h]
// One block (256 threads) per edge, one thread per channel.
// ---------------------------------------------------------------------------
__global__ void edge_scatter_kernel(const int* __restrict__ src, const int* __restrict__ dst,
                                    const float* __restrict__ xl,
                                    const float* __restrict__ ex, const float* __restrict__ den,
                                    float* __restrict__ acc)
{
    const int e = blockIdx.x;
    const int c = threadIdx.x;
    const int h = c >> 6;
    const int s = src[e], d = dst[e];
    const float alpha = ex[(size_t)e * 4 + h] / den[(size_t)d * 4 + h];
    atomicAdd(acc + (size_t)d * 256 + c, xl[(size_t)s * 256 + c] * alpha);
}

// ---------------------------------------------------------------------------
// HeteroConv mean across edge types + biases, then ELU.
// h[n][c] = elu( (acc[n][c] + b1[c] + b2[c]) * inv_cnt )
// ---------------------------------------------------------------------------
__global__ void finalize_elu_kernel(const float* __restrict__ acc,
                                    const float* __restrict__ b1,
                                    const float* __restrict__ b2,
                                    float inv_cnt, float* __restrict__ h, size_t n)
{
    size_t i = (size_t)blockIdx.x * blockDim.x + threadIdx.x;
    if (i >= n) return;
    const int c = (int)(i & 255);
    float v = acc[i] + b1[c] + (b2 ? b2[c] : 0.0f);
    v *= inv_cnt;
    h[i] = (v > 0.0f) ? v : (expf(v) - 1.0f);
}

// ---------------------------------------------------------------------------
// Host orchestration
// ---------------------------------------------------------------------------
extern "C" void kernel_launch(void* const* d_in, const int* in_sizes, int n_in,
                              void* d_out, int out_size, void* d_ws, size_t ws_size,
                              hipStream_t stream)
{
    (void)n_in; (void)out_size; (void)ws_size;
    const int IN = 384, HID = 64, HC = 256;

    // Node counts (type order: 0=table, 1=column, 2=fk_node — reference output order)
    const int nCnt[3] = { in_sizes[0] / IN, in_sizes[1] / IN, in_sizes[2] / IN };
    const float* x0[3] = { (const float*)d_in[0], (const float*)d_in[1], (const float*)d_in[2] };

    // Params flattened as JAX pytree leaves (sorted dict keys), starting at d_in[8]:
    //   layers[l] x {belongs_to,has_column,is_source_of,points_to,table_to_table}
    //             x {Wl,Wr,att,bias,bl,br}          -> 8 + l*30 + j*6 + k
    //   lin/out_lin/skip over {column,fk_node,table} x {W,b}
    auto P = [&](int i) -> const float* { return (const float*)d_in[i]; };
    auto gatBase = [&](int l, int j) { return 8 + l * 30 + j * 6; };
    const int sortedOf[3] = { 2, 0, 1 };          // table->2, column->0, fk->1 in sorted order
    const int LIN = 98, OUTL = 104, SKIP = 110;

    // Edge types in sorted-param order j = 0..4
    const int eiIdx[5] = { 4, 3, 5, 6, 7 };       // belongs_to, has_column, is_source_of, points_to, t2t
    const int srcT[5]  = { 1, 0, 1, 2, 0 };
    const int dstT[5]  = { 0, 1, 2, 1, 0 };

    // Workspace carve-out
    float* w = (float*)d_ws;
    auto carve = [&](size_t n) { float* p = w; w += n; return p; };
    float* h[3];   for (int t = 0; t < 3; ++t) h[t]   = carve((size_t)nCnt[t] * HC);
    float* acc[3]; for (int t = 0; t < 3; ++t) acc[t] = carve((size_t)nCnt[t] * HC);
    int maxN = nCnt[0]; if (nCnt[1] > maxN) maxN = nCnt[1]; if (nCnt[2] > maxN) maxN = nCnt[2];
    int maxE = 0; for (int j = 0; j < 5; ++j) { int E = in_sizes[eiIdx[j]] / 2; if (E > maxE) maxE = E; }
    float* xlb  = carve((size_t)maxN * HC);
    float* xrb  = carve((size_t)maxN * HC);
    float* lgt  = carve((size_t)maxE * 4);
    float* mbuf = carve((size_t)maxN * 4);
    float* den  = carve((size_t)maxN * 4);

    auto gemm = [&](const float* A, const float* W, const float* b, float* C,
                    int M, int N, int K, int act, int accum) {
        dim3 grid((unsigned)(((M + 31) / 32 + 7) / 8), (unsigned)(N / 64));
        gemm_wmma_f32<<<grid, 256, 0, stream>>>(A, W, b, C, M, N, K, act, accum);
    };
    auto fill = [&](float* p, size_t n, float v) {
        fill_f32<<<(unsigned)((n + 255) / 256), 256, 0, stream>>>(p, v, n);
    };

    // ---- input linear + leaky_relu(0.01) -> h (per-type, width 64) ----
    for (int t = 0; t < 3; ++t) {
        const int st = sortedOf[t];
        gemm(x0[t], P(LIN + st * 2), P(LIN + st * 2 + 1), h[t], nCnt[t], HID, IN, 1, 0);
    }

    // ---- 3 GATv2 layers ----
    int din = HID;
    for (int l = 0; l < 3; ++l) {
        for (int t = 0; t < 3; ++t) fill(acc[t], (size_t)nCnt[t] * HC, 0.0f);

        for (int j = 0; j < 5; ++j) {
            const int base = gatBase(l, j);
            const float* Wl  = P(base + 0);
            const float* Wr  = P(base + 1);
            const float* att = P(base + 2);
            const float* bl  = P(base + 4);
            const float* br  = P(base + 5);
            const int sT = srcT[j], dT = dstT[j];
            const int E = in_sizes[eiIdx[j]] / 2;
            const int* srcI = (const int*)d_in[eiIdx[j]];
            const int* dstI = srcI + E;

            gemm(h[sT], Wl, bl, xlb, nCnt[sT], HC, din, 0, 0);   // xl = x_src@Wl + bl
            gemm(h[dT], Wr, br, xrb, nCnt[dT], HC, din, 0, 0);   // xr = x_dst@Wr + br
            fill(mbuf, (size_t)nCnt[dT] * 4, -__builtin_inff());
            fill(den,  (size_t)nCnt[dT] * 4, 0.0f);

            edge_logit_kernel<<<(E + 255) / 256, 256, 0, stream>>>(
                srcI, dstI, xlb, xrb, att, lgt, mbuf, E);
            edge_expsum_kernel<<<(E * 4 + 255) / 256, 256, 0, stream>>>(
                dstI, lgt, mbuf, den, E);
            edge_scatter_kernel<<<E, 256, 0, stream>>>(
                srcI, dstI, xlb, lgt, den, acc[dT]);
        }

        // mean over edge types per dst node type (+ per-edge-type biases), then ELU
        const int b0 = gatBase(l, 0) + 3;   // belongs_to.bias  (dst=table)
        const int b1 = gatBase(l, 1) + 3;   // has_column.bias  (dst=column)
        const int b2 = gatBase(l, 2) + 3;   // is_source_of.bias(dst=fk)
        const int b3 = gatBase(l, 3) + 3;   // points_to.bias   (dst=column)
        const int b4 = gatBase(l, 4) + 3;   // t2t.bias         (dst=table)
        size_t nT = (size_t)nCnt[0] * HC, nC = (size_t)nCnt[1] * HC, nF = (size_t)nCnt[2] * HC;
        finalize_elu_kernel<<<(unsigned)((nT + 255) / 256), 256, 0, stream>>>(
            acc[0], P(b0), P(b4), 0.5f, h[0], nT);
        finalize_elu_kernel<<<(unsigned)((nC + 255) / 256), 256, 0, stream>>>(
            acc[1], P(b1), P(b3), 0.5f, h[1], nC);
        finalize_elu_kernel<<<(unsigned)((nF + 255) / 256), 256, 0, stream>>>(
            acc[2], P(b2), nullptr, 1.0f, h[2], nF);

        din = HC;
    }

    // ---- output: out = h@Wout + bout + x0@Wskip + bskip, concat table|column|fk ----
    float* out = (float*)d_out;
    size_t off = 0;
    for (int t = 0; t < 3; ++t) {
        const int st = sortedOf[t];
        gemm(h[t],  P(OUTL + st * 2), P(OUTL + st * 2 + 1), out + off, nCnt[t], 256, HC, 0, 0);
        gemm(x0[t], P(SKIP + st * 2), P(SKIP + st * 2 + 1), out + off, nCnt[t], 256, IN, 0, 1);
        off += (size_t)nCnt[t] * 256;
    }
}